// CustomMinkowskiLayerNorm_6047313953112
// MI455X (gfx1250) — compile-verified
//
#include <hip/hip_runtime.h>
#include <stdint.h>

typedef float v2f __attribute__((ext_vector_type(2)));
typedef float v4f __attribute__((ext_vector_type(4)));
typedef float v8f __attribute__((ext_vector_type(8)));

#define CCH 64
#define NBATCH 8
#define EPS_V 1e-5f
#define ROWS_PER_BLOCK 2048
#define TILE_ROWS 64
#define LDS_ROW_STRIDE 80   /* dwords: 64 data + 16 pad -> conflict-free DS reads */
#define TILE_DW (TILE_ROWS * LDS_ROW_STRIDE)   /* 5120 dwords = 20KB per buffer */
#define APPLY_VPT 4

static __device__ __forceinline__ v8f zero8() {
    v8f z;
#pragma unroll
    for (int i = 0; i < 8; ++i) z[i] = 0.0f;
    return z;
}

// ---------------------------------------------------------------------------
// Kernel 0: zero the accumulator tables (sum[8][64], sumsq[8][64])
// ---------------------------------------------------------------------------
__global__ void mlnorm_init(float* __restrict__ ws) {
    int t = threadIdx.x;
    if (t < 2 * NBATCH * CCH) ws[t] = 0.0f;
}

// ---------------------------------------------------------------------------
// Kernel 1: stats via async-to-LDS streaming + WMMA ones-reduction.
//  - 16KB tiles (64 rows x 256B) streamed with global_load_async_to_lds_b128,
//    double buffered; ASYNCcnt-paced so 2 tiles/wave stay in flight.
//  - 8 waves = 4 channel groups x 2 row-halves consume the tile from LDS and
//    accumulate sum / sum-of-squares with V_WMMA_F32_16X16X4_F32 and an
//    all-ones A (D[i][j] = sum_K B[K][j], K-permutation invariant).
//  - sorted batch ids: flush accumulators with 16 atomics on batch change;
//    boundary groups / tail rows use a scalar atomic path.
// ---------------------------------------------------------------------------
__global__ __launch_bounds__(256)
void mlnorm_stats(const float* __restrict__ X, const int* __restrict__ bidx,
                  float* __restrict__ sum, float* __restrict__ sumsq, int n) {
    __shared__ float smem[2 * TILE_DW];

    const int tid  = threadIdx.x;
    const int wave = tid >> 5;
    const int lane = tid & 31;
    const int g      = wave & 3;        // channel group (16 channels)
    const int h      = wave >> 2;       // row-half
    const int chbase = g * 16;
    const int col    = lane & 15;       // channel within group == B/D column
    const int rsel   = (lane >> 4) * 2; // lanes 0-15 rows {0,1}; 16-31 rows {2,3}

    const int s = blockIdx.x * ROWS_PER_BLOCK;
    const int e = (s + ROWS_PER_BLOCK < n) ? (s + ROWS_PER_BLOCK) : n;
    const int nfull = (e - s) / TILE_ROWS;

    const uint32_t lds0 = (uint32_t)(size_t)(&smem[0]);
    const unsigned long long xbase = (unsigned long long)(size_t)X;

    // issue the async tile load: 1024 16B chunks, 4 per thread
    auto issue = [&](int t) {
        const uint32_t lbase = lds0 + (uint32_t)((t & 1) * (TILE_DW * 4));
        const int ts = s + t * TILE_ROWS;
#pragma unroll
        for (int k = 0; k < 4; ++k) {
            int c   = tid + 256 * k;    // chunk id 0..1023
            int row = c >> 4;           // 16 chunks per 256B row
            int c4  = c & 15;
            uint32_t lds_addr = lbase +
                (uint32_t)(row * (LDS_ROW_STRIDE * 4) + c4 * 16);
            int voff = (ts + row) * (CCH * 4) + c4 * 16;  // byte offset vs X
            asm volatile("global_load_async_to_lds_b128 %0, %1, %2"
                         :: "v"(lds_addr), "v"(voff), "s"(xbase) : "memory");
        }
    };

    v2f a; a.x = 1.0f; a.y = 1.0f;      // all-ones 16x4 A matrix
    v8f dsum = zero8();
    v8f dsq  = zero8();
    int cur_b = -1;

    auto flushacc = [&]() {
        if (cur_b >= 0 && lane < 16) {
            atomicAdd(&sum  [cur_b * CCH + chbase + lane], dsum[0]);
            atomicAdd(&sumsq[cur_b * CCH + chbase + lane], dsq [0]);
        }
        dsum = zero8();
        dsq  = zero8();
    };

    if (nfull > 0) issue(0);
    if (nfull > 1) issue(1);

    for (int t = 0; t < nfull; ++t) {
        if (t + 1 < nfull) {            // next tile stays in flight
            asm volatile("s_wait_asynccnt 0x4" ::: "memory");
        } else {
            asm volatile("s_wait_asynccnt 0x0" ::: "memory");
        }
        __syncthreads();                // all waves' tile writes visible

        const float* sb = smem + (t & 1) * TILE_DW;
        const int ts = s + t * TILE_ROWS;
        const int tb0 = bidx[ts];
        const int tb1 = bidx[ts + TILE_ROWS - 1];

        if (tb0 == tb1) {               // whole tile in one batch (common)
            if (tb0 != cur_b) { flushacc(); cur_b = tb0; }
#pragma unroll
            for (int i = 0; i < 8; ++i) {
                int r0 = h * 4 + i * 8 + rsel;
                float f0 = sb[ r0      * LDS_ROW_STRIDE + chbase + col];
                float f1 = sb[(r0 + 1) * LDS_ROW_STRIDE + chbase + col];
                v2f b;  b.x  = f0;      b.y  = f1;
                v2f bq; bq.x = f0 * f0; bq.y = f1 * f1;
                dsum = __builtin_amdgcn_wmma_f32_16x16x4_f32(
                           false, a, false, b,  (short)0, dsum, false, false);
                dsq  = __builtin_amdgcn_wmma_f32_16x16x4_f32(
                           false, a, false, bq, (short)0, dsq,  false, false);
            }
        } else {                        // tile spans a segment boundary (rare)
            for (int i = 0; i < 8; ++i) {
                int rq = h * 4 + i * 8; // row within tile
                int4 i4 = *reinterpret_cast<const int4*>(bidx + ts + rq);
                if (i4.x == i4.w) {
                    if (i4.x != cur_b) { flushacc(); cur_b = i4.x; }
                    float f0 = sb[(rq + rsel)     * LDS_ROW_STRIDE + chbase + col];
                    float f1 = sb[(rq + rsel + 1) * LDS_ROW_STRIDE + chbase + col];
                    v2f b;  b.x  = f0;      b.y  = f1;
                    v2f bq; bq.x = f0 * f0; bq.y = f1 * f1;
                    dsum = __builtin_amdgcn_wmma_f32_16x16x4_f32(
                               false, a, false, b,  (short)0, dsum, false, false);
                    dsq  = __builtin_amdgcn_wmma_f32_16x16x4_f32(
                               false, a, false, bq, (short)0, dsq,  false, false);
                } else if (lane < 16) {
                    int ib[4] = {i4.x, i4.y, i4.z, i4.w};
#pragma unroll
                    for (int k = 0; k < 4; ++k) {
                        float v = sb[(rq + k) * LDS_ROW_STRIDE + chbase + lane];
                        atomicAdd(&sum  [ib[k] * CCH + chbase + lane], v);
                        atomicAdd(&sumsq[ib[k] * CCH + chbase + lane], v * v);
                    }
                }
            }
        }

        __syncthreads();                // buffer free to overwrite
        if (t + 2 < nfull) issue(t + 2);
    }

    // leftover rows (<TILE_ROWS, last block only): direct global scalar path
    if (h == 0) {
        for (int r = s + nfull * TILE_ROWS; r < e; ++r) {
            int b = bidx[r];
            if (lane < 16) {
                float v = X[(size_t)r * CCH + chbase + lane];
                atomicAdd(&sum  [b * CCH + chbase + lane], v);
                atomicAdd(&sumsq[b * CCH + chbase + lane], v * v);
            }
        }
    }

    flushacc();
}

// ---------------------------------------------------------------------------
// Kernel 2: finalize. counts via binary search on the sorted indices, then
// fold mean/var/gamma/beta into per-(b,c) scale & shift (1 FMA in pass 3).
// ---------------------------------------------------------------------------
__global__ void mlnorm_finalize(const float* __restrict__ sum,
                                const float* __restrict__ sumsq,
                                const float* __restrict__ gamma,
                                const float* __restrict__ beta,
                                const int*  __restrict__ bidx,
                                float* __restrict__ scale,
                                float* __restrict__ shift, int n) {
    int t = threadIdx.x;                // 0..511
    if (t >= NBATCH * CCH) return;
    int b = t >> 6, c = t & 63;

    auto lower = [&](int key) {
        int lo = 0, hi = n;
        while (lo < hi) {
            int mid = (lo + hi) >> 1;
            if (bidx[mid] < key) lo = mid + 1; else hi = mid;
        }
        return lo;
    };
    float cnt = (float)(lower(b + 1) - lower(b));
    if (cnt < 1.0f) cnt = 1.0f;

    float m   = sum[t] / cnt;
    float var = sumsq[t] / cnt - m * m;
    var = var < 0.0f ? 0.0f : var;
    float inv = rsqrtf(var + EPS_V);
    float sc  = gamma[c] * inv;
    scale[t] = sc;
    shift[t] = beta[c] - m * sc;
}

// ---------------------------------------------------------------------------
// Kernel 3: out = x*scale[b][c] + shift[b][c].  4 independent float4 streams
// per thread so ~2KB/wave stays in flight; nontemporal; reversed traversal to
// consume the L2-resident tail of the pass-1 feature stream first.
// ---------------------------------------------------------------------------
__global__ __launch_bounds__(256)
void mlnorm_apply(const float* __restrict__ X, const int* __restrict__ bidx,
                  const float* __restrict__ scale, const float* __restrict__ shift,
                  float* __restrict__ out, int n) {
    const long long total4 = (long long)n * (CCH / 4);
    const long long base = (long long)blockIdx.x * (256 * APPLY_VPT) + threadIdx.x;

    v4f        x[APPLY_VPT];
    int        bb[APPLY_VPT];
    long long  jj[APPLY_VPT];
    bool       ok[APPLY_VPT];

#pragma unroll
    for (int k = 0; k < APPLY_VPT; ++k) {
        long long idx = base + (long long)k * 256;
        ok[k] = idx < total4;
        long long j = total4 - 1 - idx;       // reversed traversal
        jj[k] = j;
        if (ok[k]) {
            x[k]  = __builtin_nontemporal_load(reinterpret_cast<const v4f*>(X) + j);
            bb[k] = bidx[(int)(j >> 4)];      // 16 lanes share one dword
        }
    }

#pragma unroll
    for (int k = 0; k < APPLY_VPT; ++k) {
        if (!ok[k]) continue;
        int c4 = (int)(jj[k] & 15);
        v4f sc = reinterpret_cast<const v4f*>(scale)[bb[k] * 16 + c4];
        v4f sh = reinterpret_cast<const v4f*>(shift)[bb[k] * 16 + c4];
        v4f o;
        o.x = fmaf(x[k].x, sc.x, sh.x);
        o.y = fmaf(x[k].y, sc.y, sh.y);
        o.z = fmaf(x[k].z, sc.z, sh.z);
        o.w = fmaf(x[k].w, sc.w, sh.w);
        __builtin_nontemporal_store(o, reinterpret_cast<v4f*>(out) + jj[k]);
    }
}

// ---------------------------------------------------------------------------
extern "C" void kernel_launch(void* const* d_in, const int* in_sizes, int n_in,
                              void* d_out, int out_size, void* d_ws, size_t ws_size,
                              hipStream_t stream) {
    const float* X     = (const float*)d_in[0];
    const int*   bidx  = (const int*)  d_in[1];
    const float* gamma = (const float*)d_in[2];
    const float* beta  = (const float*)d_in[3];
    float*       out   = (float*)d_out;
    const int n = in_sizes[1];

    float* ws    = (float*)d_ws;
    float* sum   = ws;                      // [8][64]
    float* sumsq = ws + NBATCH * CCH;       // [8][64]
    float* scale = ws + 2 * NBATCH * CCH;   // [8][64]
    float* shift = ws + 3 * NBATCH * CCH;   // [8][64]

    mlnorm_init<<<1, 1024, 0, stream>>>(ws);

    int nb1 = (n + ROWS_PER_BLOCK - 1) / ROWS_PER_BLOCK;
    mlnorm_stats<<<nb1, 256, 0, stream>>>(X, bidx, sum, sumsq, n);

    mlnorm_finalize<<<1, 512, 0, stream>>>(sum, sumsq, gamma, beta, bidx,
                                           scale, shift, n);

    long long total4 = (long long)n * (CCH / 4);
    int nb3 = (int)((total4 + 256LL * APPLY_VPT - 1) / (256LL * APPLY_VPT));
    mlnorm_apply<<<nb3, 256, 0, stream>>>(X, bidx, scale, shift, out, n);
}